// GCNLayer_4080218931696
// MI455X (gfx1250) — compile-verified
//
#include <hip/hip_runtime.h>

typedef float v2f __attribute__((ext_vector_type(2)));
typedef float v8f __attribute__((ext_vector_type(8)));

constexpr int DIM     = 128;   // D_IN == D_OUT == 128
constexpr int THREADS = 256;   // 8 waves of 32

// ---------------------------------------------------------------------------
// Zero the aggregation buffer (d_out) -- atomics accumulate into it.
// ---------------------------------------------------------------------------
__global__ void k_zero(float* __restrict__ p, int total) {
  int i = blockIdx.x * THREADS + threadIdx.x;
  if (i < total) p[i] = 0.0f;
}

// ---------------------------------------------------------------------------
// hw[n,:] = (h[n,:] @ W) * norm[n]   via V_WMMA_F32_16X16X4_F32
//
// One wave per 16-row tile; 8 waves / block. Weight staged in LDS swizzled
// as [kc][n][4] float4 so each lane's B-fragment is a single conflict-free
// ds_load_b64:   dword index = 4*(kc*128 + n) + 2*hi
//   lanes 0..15 (hi=0) -> banks {4n, 4n+1}; lanes 16..31 -> {4n+2, 4n+3}.
// A-fragment per ISA layout: lane l<16 holds h[row+l][k0..k0+1], lanes 16..31
// hold k0+2..k0+3 -> one 8-byte global load per chunk.
// ---------------------------------------------------------------------------
__global__ __launch_bounds__(THREADS)
void k_gemm(const float* __restrict__ h, const float* __restrict__ w,
            const float* __restrict__ norm, float* __restrict__ hw, int ntiles) {
  __shared__ float lds_w[32 * DIM * 4];   // 64 KB: [kc][n][q] = W[4*kc+q][n]

  // Cooperative swizzled stage of the 128x128 weight.
  for (int i = threadIdx.x; i < 32 * DIM; i += THREADS) {
    int kc = i >> 7;
    int n  = i & (DIM - 1);
    float4 v;
    v.x = w[(4 * kc + 0) * DIM + n];
    v.y = w[(4 * kc + 1) * DIM + n];
    v.z = w[(4 * kc + 2) * DIM + n];
    v.w = w[(4 * kc + 3) * DIM + n];
    *(float4*)(lds_w + (size_t)i * 4) = v;
  }
  __syncthreads();

  const int wave = threadIdx.x >> 5;
  const int lane = threadIdx.x & 31;
  const int hi   = lane >> 4;      // half-wave select (K pairs 0..1 vs 2..3)
  const int l    = lane & 15;
  const int tile = blockIdx.x * (THREADS / 32) + wave;
  if (tile >= ntiles) return;      // wave-uniform guard: EXEC all-ones inside

  const int rowBase   = tile * 16;
  const float* arow   = h + (size_t)(rowBase + l) * DIM + 2 * hi;
  const float* bbase  = lds_w + 4 * l + 2 * hi;

  v8f acc[8] = {};

  for (int kc = 0; kc < 32; ++kc) {
    v2f a = *(const v2f*)(arow + 4 * kc);          // global_load_b64
    const float* bk = bbase + kc * (DIM * 4);
#pragma unroll
    for (int nt = 0; nt < 8; ++nt) {
      v2f b = *(const v2f*)(bk + nt * 64);         // ds_load_b64, bank-clean
      acc[nt] = __builtin_amdgcn_wmma_f32_16x16x4_f32(
          false, a, false, b, (short)0, acc[nt], false, false);
    }
  }

  // C/D layout: VGPR j -> row rowBase + j + 8*hi, col = nt*16 + l.
  const int rb = rowBase + 8 * hi;
  float nrm[8];
#pragma unroll
  for (int j = 0; j < 8; ++j) nrm[j] = norm[rb + j];
#pragma unroll
  for (int nt = 0; nt < 8; ++nt) {
#pragma unroll
    for (int j = 0; j < 8; ++j) {
      hw[(size_t)(rb + j) * DIM + nt * 16 + l] = acc[nt][j] * nrm[j];
    }
  }
}

// ---------------------------------------------------------------------------
// Edge scatter: one wave per edge; lane c handles 4 contiguous floats.
// 32 lanes x 16 B = full 512 B row read (coalesced b128), then 4 hardware
// f32 atomic adds into the dst row. hw and out both live in the 192 MB L2.
// ---------------------------------------------------------------------------
__global__ void k_scatter(const float* __restrict__ hw, const int* __restrict__ src,
                          const int* __restrict__ dst, float* __restrict__ out, int E) {
  int gid = blockIdx.x * THREADS + threadIdx.x;
  int e = gid >> 5;
  if (e >= E) return;
  int c = (gid & 31) << 2;
  int s = src[e];
  int d = dst[e];
  const float4 v = *(const float4*)(hw + (size_t)s * DIM + c);
  float* o = out + (size_t)d * DIM + c;
  unsafeAtomicAdd(o + 0, v.x);   // global_atomic_add_f32 (no CAS loop)
  unsafeAtomicAdd(o + 1, v.y);
  unsafeAtomicAdd(o + 2, v.z);
  unsafeAtomicAdd(o + 3, v.w);
}

// ---------------------------------------------------------------------------
// out = agg * norm + bias   (in place on d_out)
// ---------------------------------------------------------------------------
__global__ void k_finalize(float* __restrict__ out, const float* __restrict__ norm,
                           const float* __restrict__ bias, int total) {
  int i = blockIdx.x * THREADS + threadIdx.x;
  if (i >= total) return;
  int n = i >> 7;           // / DIM
  int d = i & (DIM - 1);
  out[i] = out[i] * norm[n] + bias[d];
}

// ---------------------------------------------------------------------------
extern "C" void kernel_launch(void* const* d_in, const int* in_sizes, int n_in,
                              void* d_out, int out_size, void* d_ws, size_t ws_size,
                              hipStream_t stream) {
  const float* h    = (const float*)d_in[0];
  const float* norm = (const float*)d_in[1];
  const int*   src  = (const int*)d_in[2];
  const int*   dst  = (const int*)d_in[3];
  const float* w    = (const float*)d_in[4];
  const float* bias = (const float*)d_in[5];
  float* out = (float*)d_out;
  float* hw  = (float*)d_ws;            // N * 128 f32 = 51.2 MB scratch

  const int N      = in_sizes[1];       // norm has N elements
  const int E      = in_sizes[2];
  const int total  = N * DIM;
  const int ntiles = N / 16;            // 100000 / 16 = 6250, exact

  k_zero<<<(total + THREADS - 1) / THREADS, THREADS, 0, stream>>>(out, total);
  k_gemm<<<(ntiles + (THREADS / 32) - 1) / (THREADS / 32), THREADS, 0, stream>>>(
      h, w, norm, hw, ntiles);
  k_scatter<<<(int)(((size_t)E * 32 + THREADS - 1) / THREADS), THREADS, 0, stream>>>(
      hw, src, dst, out, E);
  k_finalize<<<(total + THREADS - 1) / THREADS, THREADS, 0, stream>>>(
      out, norm, bias, total);
}